// PWNPlanesLoss_64742337020675
// MI455X (gfx1250) — compile-verified
//
#include <hip/hip_runtime.h>
#include <hip/hip_bf16.h>
#include <math.h>

// ---- Problem constants (match reference) ----
#define B_  4
#define P_  4
#define H_  1088
#define W_  1920
#define N_  5000
#define K_  (B_ * P_)          // 16 planes
#define TPB 256                 // 8 wave32 waves per block
#define WAVE 32
#define NWAVES (TPB / WAVE)     // 8
#define CHUNKS ((N_ + TPB - 1) / TPB)   // 20 chunks of samples per plane

#define DELTA_COS 0.867f
#define DELTA_X   0.007f
#define DELTA_Y   0.007f

// General 3x3 inverse via adjugate (matches jnp.linalg.inv semantics).
__device__ __forceinline__ void invert3x3(const float* a, float* inv) {
    float c00 = a[4] * a[8] - a[5] * a[7];
    float c01 = a[5] * a[6] - a[3] * a[8];
    float c02 = a[3] * a[7] - a[4] * a[6];
    float det = a[0] * c00 + a[1] * c01 + a[2] * c02;
    float id  = 1.0f / det;
    inv[0] = c00 * id;
    inv[1] = (a[2] * a[7] - a[1] * a[8]) * id;
    inv[2] = (a[1] * a[5] - a[2] * a[4]) * id;
    inv[3] = c01 * id;
    inv[4] = (a[0] * a[8] - a[2] * a[6]) * id;
    inv[5] = (a[2] * a[3] - a[0] * a[5]) * id;
    inv[6] = c02 * id;
    inv[7] = (a[1] * a[6] - a[0] * a[7]) * id;
    inv[8] = (a[0] * a[4] - a[1] * a[3]) * id;
}

// Phase 1: one thread per (plane, sample). Computes the masked triangle normal
// and reduces (Sx, Sy, Sz, cnt) per (plane, chunk) into d_ws.
// Single pass thanks to:  sum_valid(1 - vn.aver_hat) = cnt - |S|^2/(|S|+1e-5).
__global__ void __launch_bounds__(TPB)
pwn_sample_kernel(const float* __restrict__ pred,        // (B,1,H,W) f32
                  const float* __restrict__ intr,        // (B,3,3)  f32
                  const int*   __restrict__ idx_y,       // (K,N,3)  i32
                  const int*   __restrict__ idx_x,       // (K,N,3)  i32
                  const int*   __restrict__ plane_batch, // (K,)     i32
                  const unsigned char* __restrict__ valid_plane, // (K,) bool
                  float4* __restrict__ partials)         // (K*CHUNKS,)
{
    const int blk   = blockIdx.x;
    const int k     = blk / CHUNKS;
    const int chunk = blk - k * CHUNKS;
    const int tid   = threadIdx.x;
    const int n     = chunk * TPB + tid;
    const int lane  = tid & (WAVE - 1);
    const int wid   = tid >> 5;

    __shared__ float sKi[9];
    __shared__ int   sB;
    __shared__ int   sValid;
    if (tid == 0) {
        int b = plane_batch[k];
        sB = b;
        sValid = (valid_plane[k] != 0) ? 1 : 0;
        float a[9];
#pragma unroll
        for (int i = 0; i < 9; ++i) a[i] = intr[b * 9 + i];
        invert3x3(a, sKi);
    }
    __syncthreads();

    const long long base = ((long long)k * N_ + n) * 3;

    // gfx1250 path: prefetch next chunk's index stream (global_prefetch_b8).
    if (n + TPB < N_) {
        __builtin_prefetch(&idx_y[base + 3LL * TPB], 0, 0);
        __builtin_prefetch(&idx_x[base + 3LL * TPB], 0, 0);
    }

    float vx = 0.0f, vy = 0.0f, vz = 0.0f, cnt = 0.0f;

    if (n < N_) {
        float gx[3], gy[3], gz[3];
        const size_t bbase = (size_t)sB * ((size_t)H_ * W_);
#pragma unroll
        for (int i = 0; i < 3; ++i) {
            int y = idx_y[base + i];
            int x = idx_x[base + i];
            float d  = pred[bbase + (size_t)y * W_ + (size_t)x];
            float ry = (float)y, rx = (float)x;           // uv = [row, col, 1]
            float c0 = fmaf(sKi[0], ry, fmaf(sKi[1], rx, sKi[2]));
            float c1 = fmaf(sKi[3], ry, fmaf(sKi[4], rx, sKi[5]));
            float c2 = fmaf(sKi[6], ry, fmaf(sKi[7], rx, sKi[8]));
            gx[i] = d * c0; gy[i] = d * c1; gz[i] = d * c2;
        }

        // 2D edge vectors in coords (0,1): d12, d13, d23
        float d12x = gx[1] - gx[0], d12y = gy[1] - gy[0];
        float d13x = gx[2] - gx[0], d13y = gy[2] - gy[0];
        float d23x = gx[2] - gx[1], d23y = gy[2] - gy[1];

        float e00 = d12x * d12x + d12y * d12y;
        float e11 = d13x * d13x + d13y * d13y;
        float e22 = d23x * d23x + d23y * d23y;
        float e01 = d12x * d13x + d12y * d13y;
        float e02 = d12x * d23x + d12y * d23y;
        float e12 = d13x * d23x + d13y * d23y;
        float n0 = sqrtf(e00), n1 = sqrtf(e11), n2 = sqrtf(e22);

        // count over the full (symmetric) 3x3 normalized-energy matrix
        int cc = 0;
        cc += (fabsf(e00 / (n0 * n0 + 1e-8f)) > DELTA_COS) ? 1 : 0;
        cc += (fabsf(e11 / (n1 * n1 + 1e-8f)) > DELTA_COS) ? 1 : 0;
        cc += (fabsf(e22 / (n2 * n2 + 1e-8f)) > DELTA_COS) ? 1 : 0;
        cc += (fabsf(e01 / (n0 * n1 + 1e-8f)) > DELTA_COS) ? 2 : 0;
        cc += (fabsf(e02 / (n0 * n2 + 1e-8f)) > DELTA_COS) ? 2 : 0;
        cc += (fabsf(e12 / (n1 * n2 + 1e-8f)) > DELTA_COS) ? 2 : 0;
        bool mask_cos = (cc > 3);

        bool mask_x = (fabsf(d12x) < DELTA_X) || (fabsf(d13x) < DELTA_X) || (fabsf(d23x) < DELTA_X);
        bool mask_y = (fabsf(d12y) < DELTA_Y) || (fabsf(d13y) < DELTA_Y) || (fabsf(d23y) < DELTA_Y);
        bool valid  = !(mask_cos || (mask_x && mask_y)) && (sValid != 0);

        // 3D triangle normal: cross(p12, p13), normalized, orientation-flipped
        float p12z = gz[1] - gz[0];
        float p13z = gz[2] - gz[0];
        float cx = d12y * p13z - p12z * d13y;
        float cy = p12z * d13x - d12x * p13z;
        float cz = d12x * d13y - d12y * d13x;
        float nn  = sqrtf(cx * cx + cy * cy + cz * cz);
        float inv = 1.0f / (nn + 1e-8f);
        cx *= inv; cy *= inv; cz *= inv;
        float orient = cx * gx[0] + cy * gy[0] + cz * gz[0];
        if (orient > 0.0f) { cx = -cx; cy = -cy; cz = -cz; }

        if (valid) { vx = cx; vy = cy; vz = cz; cnt = 1.0f; }
    }

    // ---- Deterministic reduction, wave32-shaped ----
    // Level 1: intra-wave shuffle tree (no LDS memory, no barriers).
#pragma unroll
    for (int off = WAVE / 2; off > 0; off >>= 1) {
        vx  += __shfl_down(vx,  off, WAVE);
        vy  += __shfl_down(vy,  off, WAVE);
        vz  += __shfl_down(vz,  off, WAVE);
        cnt += __shfl_down(cnt, off, WAVE);
    }

    // Level 2: 8 warp partials through LDS, single barrier.
    __shared__ float4 warpSums[NWAVES];
    if (lane == 0) warpSums[wid] = make_float4(vx, vy, vz, cnt);
    __syncthreads();

    // Level 3: wave 0 combines the 8 partials with a fixed shuffle tree.
    if (wid == 0) {
        float4 p = (lane < NWAVES) ? warpSums[lane]
                                   : make_float4(0.0f, 0.0f, 0.0f, 0.0f);
        float sx = p.x, sy = p.y, sz = p.z, sc = p.w;
#pragma unroll
        for (int off = NWAVES / 2; off > 0; off >>= 1) {
            sx += __shfl_down(sx, off, WAVE);
            sy += __shfl_down(sy, off, WAVE);
            sz += __shfl_down(sz, off, WAVE);
            sc += __shfl_down(sc, off, WAVE);
        }
        if (lane == 0) partials[blk] = make_float4(sx, sy, sz, sc);
    }
}

// Phase 2: single wave32 block. Threads 0..15 finish their plane's sum in a
// fixed order (deterministic), thread 0 combines planes and writes the loss.
__global__ void __launch_bounds__(32)
pwn_finalize_kernel(const float4* __restrict__ partials, float* __restrict__ out)
{
    __shared__ float sLoss[K_];
    __shared__ float sCnt[K_];
    const int t = threadIdx.x;
    if (t < K_) {
        float sx = 0.0f, sy = 0.0f, sz = 0.0f, c = 0.0f;
#pragma unroll
        for (int i = 0; i < CHUNKS; ++i) {
            float4 p = partials[t * CHUNKS + i];
            sx += p.x; sy += p.y; sz += p.z; c += p.w;
        }
        float ss = sx * sx + sy * sy + sz * sz;
        // loss_k = cnt - S.S/(|S| + 1e-5)  ==  sum_valid(1 - vn . aver_hat)
        float lk = c - ss / (sqrtf(ss) + 1e-5f);
        bool ok  = (c >= 2.0f);
        sLoss[t] = ok ? lk : 0.0f;
        sCnt[t]  = ok ? c  : 0.0f;
    }
    __syncthreads();
    if (t == 0) {
        float ls = 0.0f, vnum = 0.0f;
#pragma unroll
        for (int kk = 0; kk < K_; ++kk) { ls += sLoss[kk]; vnum += sCnt[kk]; }
        out[0] = ls / (vnum + 1e-6f);   // * LOSS_WEIGHT (1.0)
    }
}

extern "C" void kernel_launch(void* const* d_in, const int* in_sizes, int n_in,
                              void* d_out, int out_size, void* d_ws, size_t ws_size,
                              hipStream_t stream) {
    (void)in_sizes; (void)n_in; (void)out_size; (void)ws_size;
    const float* pred = (const float*)d_in[0];            // (B,1,H,W) f32
    const float* intr = (const float*)d_in[1];            // (B,3,3)  f32
    const int*   iy   = (const int*)d_in[2];              // (K,N,3)  i32
    const int*   ix   = (const int*)d_in[3];              // (K,N,3)  i32
    const int*   pb   = (const int*)d_in[4];              // (K,)     i32
    const unsigned char* vp = (const unsigned char*)d_in[5]; // (K,)  bool
    float4* partials = (float4*)d_ws;                     // K*CHUNKS float4 = 5 KB

    pwn_sample_kernel<<<dim3(K_ * CHUNKS), dim3(TPB), 0, stream>>>(
        pred, intr, iy, ix, pb, vp, partials);
    pwn_finalize_kernel<<<dim3(1), dim3(32), 0, stream>>>(partials, (float*)d_out);
}